// GenModel_81381040325247
// MI455X (gfx1250) — compile-verified
//
#include <hip/hip_runtime.h>

typedef float v2f __attribute__((ext_vector_type(2)));
typedef float v8f __attribute__((ext_vector_type(8)));

// D = A(16x4) * B(4x16) + C, fp32, wave32
#define WMMA_F32(A, B, C) \
  __builtin_amdgcn_wmma_f32_16x16x4_f32(false, (A), false, (B), (short)0, (C), false, false)

constexpr int kN = 4096;          // particles
constexpr int kRK = 64;           // ODE steps
constexpr int kEM = kN - 1;       // EM steps (4095)
constexpr float kDt = 1.0f / 64.0f;

__device__ __forceinline__ float fast_tanh(float x) {
  // tanh(x) = 1 - 2/(exp(2x)+1), via v_exp_f32 / v_rcp_f32
  float e = __builtin_amdgcn_exp2f(x * 2.8853900817779268f);   // 2*log2(e)
  float r = __builtin_amdgcn_rcpf(e + 1.0f);
  return __builtin_fmaf(-2.0f, r, 1.0f);
}
__device__ __forceinline__ float fast_sigmoid(float x) {
  float e = __builtin_amdgcn_exp2f(x * -1.4426950408889634f);  // -log2(e)
  return __builtin_amdgcn_rcpf(1.0f + e);
}

// Row permutation so layer-1 C output lands directly in layer-2 B layout:
// tile0 row M -> hidden H0(M) = [0,1,4,5,8,9,12,13 | 2,3,6,7,10,11,14,15]
__device__ __forceinline__ int permH(int m) {
  return 4 * ((m & 7) >> 1) + (m & 1) + 2 * (m >> 3);
}

struct MlpA {
  v2f a1t0;   // layer1 A-tile (row-permuted), hidden {0..15}, bias via K=2
  v2f a1t1;   // layer1 A-tile (row-permuted), hidden {16..23}, other rows zero
  v2f a2[6];  // layer2 A chunks over K=24 (rows 0..1 = W2^T, rest zero)
};

// Result: .x = out.x per particle (lanes 0-15), .y = out.y per particle.
__device__ __forceinline__ v2f mlp_eval(const MlpA& W, float zx, float zy, bool lo) {
  v2f b1;
  b1.x = lo ? zx : 1.0f;   // K=0: x   | K=2: ones (bias feed)
  b1.y = lo ? zy : 0.0f;   // K=1: y   | K=3: zero
  v8f zero = {0.f, 0.f, 0.f, 0.f, 0.f, 0.f, 0.f, 0.f};
  v8f t0 = WMMA_F32(W.a1t0, b1, zero);   // hidden 0..15 (permuted rows)
  v8f t1 = WMMA_F32(W.a1t1, b1, zero);   // hidden 16..23 in VGPRs 0..3
#pragma unroll
  for (int i = 0; i < 8; ++i) t0[i] = fast_tanh(t0[i]);
#pragma unroll
  for (int i = 0; i < 4; ++i) t1[i] = fast_tanh(t1[i]);

  // Row permutation makes C VGPR pairs the layer-2 B operands directly.
  // All 6 chunk-WMMAs independent (depth 1), reduced by a short add tree.
  v8f c0 = WMMA_F32(W.a2[0], __builtin_shufflevector(t0, t0, 0, 1), zero);
  v8f c1 = WMMA_F32(W.a2[1], __builtin_shufflevector(t0, t0, 2, 3), zero);
  v8f c2 = WMMA_F32(W.a2[2], __builtin_shufflevector(t0, t0, 4, 5), zero);
  v8f c3 = WMMA_F32(W.a2[3], __builtin_shufflevector(t0, t0, 6, 7), zero);
  v8f c4 = WMMA_F32(W.a2[4], __builtin_shufflevector(t1, t1, 0, 1), zero);
  v8f c5 = WMMA_F32(W.a2[5], __builtin_shufflevector(t1, t1, 2, 3), zero);
  v2f r;
  r.x = ((c0[0] + c1[0]) + (c2[0] + c3[0])) + (c4[0] + c5[0]);
  r.y = ((c0[1] + c1[1]) + (c2[1] + c3[1])) + (c4[1] + c5[1]);
  return r;
}

__global__ __launch_bounds__(256) void sde_wmma_kernel(
    const float* __restrict__ Wf1, const float* __restrict__ bf1,
    const float* __restrict__ Wf2, const float* __restrict__ bf2,
    const float* __restrict__ Wg1, const float* __restrict__ bg1,
    const float* __restrict__ Wg2, const float* __restrict__ bg2,
    const float* __restrict__ term_loc, const float* __restrict__ log_term_std,
    const float* __restrict__ noise_eps, const float* __restrict__ dW,
    float* __restrict__ out) {
  const int lane = (int)(threadIdx.x & 31);
  const int m = lane & 15;
  const bool lo = lane < 16;
  const int half = lo ? 0 : 1;
  const int wave = (int)((blockIdx.x * blockDim.x + threadIdx.x) >> 5);
  const int base = wave * 16;  // 16 particles per wave, 256 waves total

  // ---- Build wave-resident A matrices (reused across every step) ----
  MlpA F, G;
  const int h0 = permH(m);
  F.a1t0.x = lo ? Wf1[h0]      : bf1[h0];
  F.a1t0.y = lo ? Wf1[24 + h0] : 0.0f;
  G.a1t0.x = lo ? Wg1[h0]      : bg1[h0];
  G.a1t0.y = lo ? Wg1[24 + h0] : 0.0f;
  const bool t1ok = (m & 7) < 4;
  int h1 = 16 + 4 * ((m & 7) >> 1) + (m & 1) + 2 * (m >> 3);
  if (!t1ok) h1 = 0;  // clamp for safe loads; value masked below
  F.a1t1.x = t1ok ? (lo ? Wf1[h1] : bf1[h1]) : 0.0f;
  F.a1t1.y = (t1ok && lo) ? Wf1[24 + h1] : 0.0f;
  G.a1t1.x = t1ok ? (lo ? Wg1[h1] : bg1[h1]) : 0.0f;
  G.a1t1.y = (t1ok && lo) ? Wg1[24 + h1] : 0.0f;
#pragma unroll
  for (int c = 0; c < 6; ++c) {
    F.a2[c].x = (m < 2) ? Wf2[(4 * c + 0 + 2 * half) * 2 + m] : 0.0f;
    F.a2[c].y = (m < 2) ? Wf2[(4 * c + 1 + 2 * half) * 2 + m] : 0.0f;
    G.a2[c].x = (m == 0) ? Wg2[4 * c + 0 + 2 * half] : 0.0f;
    G.a2[c].y = (m == 0) ? Wg2[4 * c + 1 + 2 * half] : 0.0f;
  }
  const float bf2x = bf2[0], bf2y = bf2[1], bg2s = bg2[0];

  // ---- z_term ----
  const int n = base + m;  // upper lanes mirror lower (they never store)
  const int p4 = n & 3;
  const float stdv = __builtin_amdgcn_exp2f(log_term_std[p4] * 1.4426950408889634f);
  float zx = term_loc[p4 * 2 + 0] + stdv * noise_eps[n * 2 + 0];
  float zy = term_loc[p4 * 2 + 1] + stdv * noise_eps[n * 2 + 1];

  // ---- RK4 backward integration: dz/dt = -f(z) ----
#pragma unroll 1
  for (int s = 0; s < kRK; ++s) {
    v2f o1 = mlp_eval(F, zx, zy, lo);
    float k1x = -(o1.x + bf2x), k1y = -(o1.y + bf2y);
    v2f o2 = mlp_eval(F, zx + 0.5f * kDt * k1x, zy + 0.5f * kDt * k1y, lo);
    float k2x = -(o2.x + bf2x), k2y = -(o2.y + bf2y);
    v2f o3 = mlp_eval(F, zx + 0.5f * kDt * k2x, zy + 0.5f * kDt * k2y, lo);
    float k3x = -(o3.x + bf2x), k3y = -(o3.y + bf2y);
    v2f o4 = mlp_eval(F, zx + kDt * k3x, zy + kDt * k3y, lo);
    float k4x = -(o4.x + bf2x), k4y = -(o4.y + bf2y);
    zx += kDt * (1.0f / 6.0f) * (k1x + 2.0f * k2x + 2.0f * k3x + k4x);
    zy += kDt * (1.0f / 6.0f) * (k1y + 2.0f * k2y + 2.0f * k3y + k4y);
  }

  float* __restrict__ z_init_out = out;                                          // (N,2)
  float* __restrict__ z_forw     = out + (size_t)2 * kN;                         // (N,N,2)
  float* __restrict__ z_samp     = out + (size_t)2 * kN + (size_t)2 * kN * kN;   // (N,2)

  if (lo) {
    z_init_out[2 * n + 0] = zx; z_init_out[2 * n + 1] = zy;
    z_forw[2 * n + 0] = zx;     z_forw[2 * n + 1] = zy;   // row 0 = z_init
    if (n == 0) { z_samp[0] = zx; z_samp[1] = zy; }       // z_samples[0]
  }

  // ---- Euler-Maruyama forward scan ----
  const float dts = 1.0f / (float)kEM;
  const float sq = sqrtf(dts);
  // Unconditional loads: upper lanes duplicate lower-lane addresses (same
  // 128B line, no extra traffic) -> no saveexec region in the hot loop.
  float dwx = __builtin_nontemporal_load(&dW[(size_t)n * 2 + 0]);
  float dwy = __builtin_nontemporal_load(&dW[(size_t)n * 2 + 1]);

#pragma unroll 1
  for (int t = 0; t < kEM; ++t) {
    const float cdx = dwx, cdy = dwy;
    {  // deep prefetch (global_prefetch_b8) + one-step-ahead pipeline
      int tp = (t + 8 < kEM) ? (t + 8) : (kEM - 1);
      __builtin_prefetch(&dW[(size_t)tp * (2 * kN) + (size_t)n * 2], 0, 1);
      int tn = (t + 1 < kEM) ? (t + 1) : t;
      size_t off = (size_t)tn * (2 * kN) + (size_t)n * 2;
      dwx = __builtin_nontemporal_load(&dW[off]);
      dwy = __builtin_nontemporal_load(&dW[off + 1]);
    }
    v2f of = mlp_eval(F, zx, zy, lo);
    v2f og = mlp_eval(G, zx, zy, lo);
    const float fx = of.x + bf2x, fy = of.y + bf2y;
    const float gv = 0.3f * fast_sigmoid(og.x + bg2s);
    zx = zx + fx * dts + gv * sq * cdx;
    zy = zy + fy * dts + gv * sq * cdy;
    if (lo) {
      size_t o = (size_t)(t + 1) * (2 * kN) + (size_t)n * 2;
      __builtin_nontemporal_store(zx, &z_forw[o]);       // streamed, never re-read
      __builtin_nontemporal_store(zy, &z_forw[o + 1]);
      if ((t + 1) - base == m) {  // diagonal sample z_samples[t+1]
        z_samp[2 * (t + 1) + 0] = zx;
        z_samp[2 * (t + 1) + 1] = zy;
      }
    }
  }
}

extern "C" void kernel_launch(void* const* d_in, const int* in_sizes, int n_in,
                              void* d_out, int out_size, void* d_ws, size_t ws_size,
                              hipStream_t stream) {
  // setup_inputs() order: N, Wf1, bf1, Wf2, bf2, Wg1, bg1, Wg2, bg2,
  //                       term_loc, log_term_std, noise_eps, dW
  const float* Wf1 = (const float*)d_in[1];
  const float* bf1 = (const float*)d_in[2];
  const float* Wf2 = (const float*)d_in[3];
  const float* bf2 = (const float*)d_in[4];
  const float* Wg1 = (const float*)d_in[5];
  const float* bg1 = (const float*)d_in[6];
  const float* Wg2 = (const float*)d_in[7];
  const float* bg2 = (const float*)d_in[8];
  const float* term_loc = (const float*)d_in[9];
  const float* log_term_std = (const float*)d_in[10];
  const float* noise_eps = (const float*)d_in[11];
  const float* dW = (const float*)d_in[12];

  // 4096 particles / 16 per wave = 256 waves; 8 waves (256 thr) per block.
  sde_wmma_kernel<<<32, 256, 0, stream>>>(Wf1, bf1, Wf2, bf2, Wg1, bg1, Wg2, bg2,
                                          term_loc, log_term_std, noise_eps, dW,
                                          (float*)d_out);
}